// PytorchMambaPrefillSSM_28020366639746
// MI455X (gfx1250) — compile-verified
//
#include <hip/hip_runtime.h>
#include <math.h>

// Problem constants (from reference): B=2, D_IN=4096, L=4096, N=16, R=128
#define BDIM    2
#define DIN     4096
#define LLEN    4096
#define NST     16
#define RDIM    128
#define KROWS   160          // R + 2N rows of x_proj_w
#define SCHUNK  32           // scan chunks
#define TCHUNK  128          // L / SCHUNK

typedef __attribute__((ext_vector_type(2))) float v2f;
typedef __attribute__((ext_vector_type(8))) float v8f;

static __device__ __forceinline__ float softplusf(float v) {
    // stable softplus, matches jax.nn.softplus to fp32 accuracy
    return v > 20.f ? v : log1pf(expf(v));
}

// ---------------------------------------------------------------------------
// Kernel 1: xd[b,k,l] = sum_d w[k,d] * x[b,d,l]   (160 x 4096) @ (4096 x L)
// Block = 320 threads = 10 waves; each wave owns one 16-row k-tile at a fixed
// 16-column l-tile. x tile (32d x 16l) is staged transposed in LDS and shared
// by all 10 waves. f32 WMMA 16x16x4, K unrolled by 8 (d-chunk of 32).
// ---------------------------------------------------------------------------
__global__ __launch_bounds__(320)
void gemm_xproj(const float* __restrict__ x, const float* __restrict__ w,
                float* __restrict__ xd)
{
    __shared__ float xs[16 * 32];          // [l_col][d_row] transposed tile
    const int tid  = threadIdx.x;
    const int l0   = blockIdx.x * 16;
    const int b    = blockIdx.y;
    const int lane = tid & 31;
    const int wid  = tid >> 5;             // k-tile 0..9
    const int m    = lane & 15;            // row within tile (A) / col (B,C)
    const int hi   = lane >> 4;
    const int koff = hi * 2;               // K sub-offset per A/B f32 layout

    const float* xb = x + (size_t)b * DIN * LLEN;
    v8f acc = {0.f,0.f,0.f,0.f,0.f,0.f,0.f,0.f};

    for (int d0 = 0; d0 < DIN; d0 += 32) {
        if (tid < 256) {
            #pragma unroll
            for (int j = 0; j < 2; ++j) {
                int idx = tid + 256 * j;           // 512 elements
                int row = idx >> 4;                // d offset 0..31
                int col = idx & 15;                // l offset (coalesced)
                xs[col * 32 + row] = xb[(size_t)(d0 + row) * LLEN + l0 + col];
            }
        }
        __syncthreads();
        const float* wrow = w + (size_t)(wid * 16 + m) * DIN + d0;
        #pragma unroll
        for (int c = 0; c < 8; ++c) {
            const int k = 4 * c + koff;
            v2f af; af.x = wrow[k];            af.y = wrow[k + 1];
            v2f bf; bf.x = xs[m * 32 + k];     bf.y = xs[m * 32 + k + 1];
            acc = __builtin_amdgcn_wmma_f32_16x16x4_f32(
                      false, af, false, bf, (short)0, acc, false, false);
        }
        __syncthreads();
    }
    // C/D layout: VGPR r -> M = r + 8*hi, N = m
    float* xdb = xd + ((size_t)b * KROWS + wid * 16) * LLEN + l0 + m;
    #pragma unroll
    for (int r = 0; r < 8; ++r)
        xdb[(size_t)(r + hi * 8) * LLEN] = acc[r];
}

// ---------------------------------------------------------------------------
// Kernel 2: delta[b,d,l] = softplus( dtw(4096x128) @ xd[b,0:128,l] + bias[d] )
// Block = 256 threads = 8 waves; 8 d-tiles share one LDS-staged (16l x 128r)
// transposed B tile. K=128 fully unrolled -> 32 WMMAs per tile.
// ---------------------------------------------------------------------------
__global__ __launch_bounds__(256)
void gemm_dt(const float* __restrict__ xd, const float* __restrict__ dtw,
             const float* __restrict__ dtb, float* __restrict__ delta)
{
    __shared__ float bs[16 * RDIM];        // [l_col][r] transposed tile
    const int tid  = threadIdx.x;
    const int l0   = blockIdx.x * 16;
    const int dg   = blockIdx.y;           // 0..31 (groups of 8 d-tiles)
    const int b    = blockIdx.z;
    const int lane = tid & 31;
    const int wid  = tid >> 5;             // 0..7
    const int m    = lane & 15;
    const int hi   = lane >> 4;
    const int koff = hi * 2;
    const int d0   = (dg * 8 + wid) * 16;

    #pragma unroll
    for (int j = 0; j < 8; ++j) {
        int idx = tid + 256 * j;           // 2048 elements
        int col = idx & 15;                // l (coalesced)
        int r   = idx >> 4;                // 0..127
        bs[col * RDIM + r] = xd[((size_t)b * KROWS + r) * LLEN + l0 + col];
    }
    __syncthreads();

    const float* arow = dtw + (size_t)(d0 + m) * RDIM;
    v8f acc = {0.f,0.f,0.f,0.f,0.f,0.f,0.f,0.f};
    #pragma unroll
    for (int c = 0; c < 32; ++c) {
        const int k = 4 * c + koff;
        v2f af; af.x = arow[k];             af.y = arow[k + 1];
        v2f bf; bf.x = bs[m * RDIM + k];    bf.y = bs[m * RDIM + k + 1];
        acc = __builtin_amdgcn_wmma_f32_16x16x4_f32(
                  false, af, false, bf, (short)0, acc, false, false);
    }
    #pragma unroll
    for (int r = 0; r < 8; ++r) {
        const int drow = d0 + r + hi * 8;
        float v = acc[r] + dtb[drow];
        delta[((size_t)b * DIN + drow) * LLEN + l0 + m] = softplusf(v);
    }
}

// ---------------------------------------------------------------------------
// Scan pass 1: per (b, chunk s, channel d) compute local state h (zero init)
// and cumulative decay prod(exp(delta*A)) over T=128 steps.
// B_t rows (shared by all channels) staged in LDS.
// ---------------------------------------------------------------------------
__global__ __launch_bounds__(256)
void scan_pass1(const float* __restrict__ delta, const float* __restrict__ xd,
                const float* __restrict__ x, const float* __restrict__ A_log,
                float* __restrict__ hloc, float* __restrict__ decay)
{
    __shared__ float Bt[NST * TCHUNK];     // 8 KB
    const int tid = threadIdx.x;
    const int d   = blockIdx.x * 256 + tid;
    const int s   = blockIdx.y;
    const int b   = blockIdx.z;

    #pragma unroll
    for (int j = 0; j < 8; ++j) {
        int idx = tid + 256 * j;           // n = idx>>7, t = idx&127
        int n = idx >> 7, t = idx & 127;
        Bt[idx] = xd[((size_t)b * KROWS + RDIM + n) * LLEN + s * TCHUNK + t];
    }
    __syncthreads();

    float a[NST], h[NST], dc[NST];
    const float* al = A_log + (size_t)d * NST;
    #pragma unroll
    for (int n = 0; n < NST; ++n) { a[n] = -expf(al[n]); h[n] = 0.f; dc[n] = 1.f; }

    const size_t ro = ((size_t)b * DIN + d) * LLEN + (size_t)s * TCHUNK;
    const float4* dl4 = (const float4*)(delta + ro);
    const float4* xx4 = (const float4*)(x + ro);
    for (int t4 = 0; t4 < TCHUNK / 4; ++t4) {
        float4 dv = dl4[t4], xv = xx4[t4];
        float dq[4] = {dv.x, dv.y, dv.z, dv.w};
        float uq[4] = {xv.x, xv.y, xv.z, xv.w};
        #pragma unroll
        for (int q = 0; q < 4; ++q) {
            const float du = dq[q], su = dq[q] * uq[q];
            const int t = t4 * 4 + q;
            #pragma unroll
            for (int n = 0; n < NST; ++n) {
                float e = expf(du * a[n]);
                dc[n] *= e;
                h[n] = h[n] * e + su * Bt[n * TCHUNK + t];
            }
        }
    }
    const size_t o = (((size_t)b * SCHUNK + s) * DIN + d) * NST;
    #pragma unroll
    for (int n = 0; n < NST; ++n) { hloc[o + n] = h[n]; decay[o + n] = dc[n]; }
}

// ---------------------------------------------------------------------------
// Combine: short sequential pass over the 32 chunks per channel to produce
// each chunk's entry state. 8192 threads, 32 iterations.
// ---------------------------------------------------------------------------
__global__ __launch_bounds__(256)
void scan_combine(const float* __restrict__ hloc, const float* __restrict__ decay,
                  float* __restrict__ hin)
{
    const int idx = blockIdx.x * 256 + threadIdx.x;   // b*DIN + d
    const int b = idx / DIN, d = idx % DIN;
    float H[NST];
    #pragma unroll
    for (int n = 0; n < NST; ++n) H[n] = 0.f;
    for (int s = 0; s < SCHUNK; ++s) {
        const size_t o = (((size_t)b * SCHUNK + s) * DIN + d) * NST;
        #pragma unroll
        for (int n = 0; n < NST; ++n) hin[o + n] = H[n];
        #pragma unroll
        for (int n = 0; n < NST; ++n) H[n] = H[n] * decay[o + n] + hloc[o + n];
    }
}

// ---------------------------------------------------------------------------
// Scan pass 2: rerun each chunk with the correct entry state; emit
// y = sum_n h[n]*C_t[n] + x*D. B and C rows staged in LDS (16 KB).
// ---------------------------------------------------------------------------
__global__ __launch_bounds__(256)
void scan_pass2(const float* __restrict__ delta, const float* __restrict__ xd,
                const float* __restrict__ x, const float* __restrict__ A_log,
                const float* __restrict__ Dvec, const float* __restrict__ hin,
                float* __restrict__ out)
{
    __shared__ float Bt[NST * TCHUNK];
    __shared__ float Ct[NST * TCHUNK];
    const int tid = threadIdx.x;
    const int d   = blockIdx.x * 256 + tid;
    const int s   = blockIdx.y;
    const int b   = blockIdx.z;

    #pragma unroll
    for (int j = 0; j < 8; ++j) {
        int idx = tid + 256 * j;
        int n = idx >> 7, t = idx & 127;
        Bt[idx] = xd[((size_t)b * KROWS + RDIM + n)       * LLEN + s * TCHUNK + t];
        Ct[idx] = xd[((size_t)b * KROWS + RDIM + NST + n) * LLEN + s * TCHUNK + t];
    }
    __syncthreads();

    float a[NST], h[NST];
    const float* al = A_log + (size_t)d * NST;
    const size_t ho = (((size_t)b * SCHUNK + s) * DIN + d) * NST;
    #pragma unroll
    for (int n = 0; n < NST; ++n) { a[n] = -expf(al[n]); h[n] = hin[ho + n]; }
    const float dd = Dvec[d];

    const size_t ro = ((size_t)b * DIN + d) * LLEN + (size_t)s * TCHUNK;
    const float4* dl4 = (const float4*)(delta + ro);
    const float4* xx4 = (const float4*)(x + ro);
    float4* out4 = (float4*)(out + ro);
    for (int t4 = 0; t4 < TCHUNK / 4; ++t4) {
        float4 dv = dl4[t4], xv = xx4[t4];
        float dq[4] = {dv.x, dv.y, dv.z, dv.w};
        float uq[4] = {xv.x, xv.y, xv.z, xv.w};
        float yy[4];
        #pragma unroll
        for (int q = 0; q < 4; ++q) {
            const float du = dq[q], u = uq[q], su = du * u;
            const int t = t4 * 4 + q;
            float y = 0.f;
            #pragma unroll
            for (int n = 0; n < NST; ++n) {
                float e = expf(du * a[n]);
                h[n] = h[n] * e + su * Bt[n * TCHUNK + t];
                y += h[n] * Ct[n * TCHUNK + t];
            }
            yy[q] = y + u * dd;
        }
        float4 yv; yv.x = yy[0]; yv.y = yy[1]; yv.z = yy[2]; yv.w = yy[3];
        out4[t4] = yv;
    }
}

// ---------------------------------------------------------------------------
extern "C" void kernel_launch(void* const* d_in, const int* in_sizes, int n_in,
                              void* d_out, int out_size, void* d_ws, size_t ws_size,
                              hipStream_t stream)
{
    const float* x     = (const float*)d_in[0];
    const float* A_log = (const float*)d_in[1];
    const float* Dvec  = (const float*)d_in[2];
    const float* xpw   = (const float*)d_in[3];
    const float* dtw   = (const float*)d_in[4];
    const float* dtb   = (const float*)d_in[5];
    float* out = (float*)d_out;

    // workspace layout (floats): ~190 MB total
    float* ws    = (float*)d_ws;
    float* xd    = ws;                                         // B*160*L
    float* delta = xd    + (size_t)BDIM * KROWS * LLEN;        // B*D*L
    float* hloc  = delta + (size_t)BDIM * DIN * LLEN;          // B*S*D*16
    float* decay = hloc  + (size_t)BDIM * SCHUNK * DIN * NST;
    float* hin   = decay + (size_t)BDIM * SCHUNK * DIN * NST;

    gemm_xproj <<<dim3(LLEN / 16, BDIM),            320, 0, stream>>>(x, xpw, xd);
    gemm_dt    <<<dim3(LLEN / 16, DIN / 128, BDIM), 256, 0, stream>>>(xd, dtw, dtb, delta);
    scan_pass1 <<<dim3(DIN / 256, SCHUNK, BDIM),    256, 0, stream>>>(delta, xd, x, A_log, hloc, decay);
    scan_combine<<<dim3((BDIM * DIN) / 256),        256, 0, stream>>>(hloc, decay, hin);
    scan_pass2 <<<dim3(DIN / 256, SCHUNK, BDIM),    256, 0, stream>>>(delta, xd, x, A_log, Dvec, hin, out);
}